// VectorQuantizer_8392366096890
// MI455X (gfx1250) — compile-verified
//
#include <hip/hip_runtime.h>

typedef __attribute__((ext_vector_type(2))) float v2f;
typedef __attribute__((ext_vector_type(8))) float v8f;

#define N_ROWS    65536      // B*D*H*W = 4*8*32*64
#define K_CODES   8192
#define CDIM      4
#define CH_STRIDE 16384      // D*H*W
#define B_STRIDE  65536      // C*CH_STRIDE
#define CHUNK     2048       // codes staged in LDS at a time (32KB + 8KB e2)
#define TILES_PER_CHUNK (CHUNK / 16)
#define NCHUNK    (K_CODES / CHUNK)
#define WAVES_PER_WG 8
#define RT_PER_WAVE  2       // row-tiles (16 rows each) per wave
#define ROWS_PER_WG  (WAVES_PER_WG * RT_PER_WAVE * 16)   // 256

// d_out layout (flat, return order): z_q_st [262144] | loss [1] | indices [65536]
#define OUT_LOSS_OFF (N_ROWS * CDIM)
#define OUT_IDX_OFF  (N_ROWS * CDIM + 1)

__global__ __launch_bounds__(256) void vq_main(const float* __restrict__ z,
                                               const float* __restrict__ cb,
                                               float* __restrict__ out,
                                               float* __restrict__ partial)
{
    __shared__ float lds_cb[CHUNK * 4];   // codebook chunk, row-major (code, c)
    __shared__ float lds_e2[CHUNK];       // per-code squared norms
    __shared__ float lds_red[256];        // loss reduction

    const int tid   = threadIdx.x;
    const int wave  = tid >> 5;
    const int lane  = tid & 31;
    const int m16   = lane & 15;           // row (A) / column (B,D) within tile
    const int khalf = (lane >> 4) * 2;     // K components held by this half-wave

    // ---- A matrices: 16x4 f32, A = -2*z  (so D = -2 z.e; dist = D + ||e||^2) ----
    // z_flat[r][k] = z[(r>>14)*B_STRIDE + k*CH_STRIDE + (r & (CH_STRIDE-1))]
    const int wave_row_base = blockIdx.x * ROWS_PER_WG + wave * (RT_PER_WAVE * 16);
    v2f a[RT_PER_WAVE];
#pragma unroll
    for (int t = 0; t < RT_PER_WAVE; ++t) {
        int r  = wave_row_base + t * 16 + m16;
        int bb = r >> 14;
        int mm = r & (CH_STRIDE - 1);
        const float* zp = z + bb * B_STRIDE + mm;
        a[t].x = -2.0f * zp[(khalf + 0) * CH_STRIDE];
        a[t].y = -2.0f * zp[(khalf + 1) * CH_STRIDE];
    }

    float bestd[RT_PER_WAVE][8];
    int   besti[RT_PER_WAVE][8];
#pragma unroll
    for (int t = 0; t < RT_PER_WAVE; ++t)
#pragma unroll
        for (int v = 0; v < 8; ++v) { bestd[t][v] = 3.402823466e38f; besti[t][v] = 0; }

    const v8f czero = {};

    for (int ch = 0; ch < NCHUNK; ++ch) {
        // cooperative chunk load: 2048 codes * float4, coalesced
        const float4* src = (const float4*)cb + ch * CHUNK;
#pragma unroll
        for (int j = 0; j < CHUNK / 256; ++j) {
            int i = j * 256 + tid;
            float4 e = src[i];
            ((float4*)lds_cb)[i] = e;
            lds_e2[i] = e.x * e.x + e.y * e.y + e.z * e.z + e.w * e.w;
        }
        __syncthreads();

        const int gbase = ch * CHUNK;
        for (int tt = 0; tt < TILES_PER_CHUNK; ++tt) {
            int   n    = tt * 16 + m16;                       // chunk-local code (column)
            v2f   bv   = *(const v2f*)&lds_cb[n * 4 + khalf]; // B: 4x16, ds_load_b64
            float e2   = lds_e2[n];
            int   gidx = gbase + n;
#pragma unroll
            for (int t = 0; t < RT_PER_WAVE; ++t) {
                v8f d = __builtin_amdgcn_wmma_f32_16x16x4_f32(
                            false, a[t], false, bv, (short)0, czero, false, false);
#pragma unroll
                for (int v = 0; v < 8; ++v) {
                    float dist = d[v] + e2;
                    bool take  = dist < bestd[t][v];   // strict <: gidx increases => first-min kept
                    bestd[t][v] = take ? dist : bestd[t][v];
                    besti[t][v] = take ? gidx : besti[t][v];
                }
            }
        }
        __syncthreads();
    }

    // ---- cross-lane argmin: each D row spans 16 lanes (0-15: M=v, 16-31: M=v+8) ----
#pragma unroll
    for (int t = 0; t < RT_PER_WAVE; ++t)
#pragma unroll
        for (int v = 0; v < 8; ++v)
#pragma unroll
            for (int mask = 1; mask <= 8; mask <<= 1) {
                float od = __shfl_xor(bestd[t][v], mask, 32);
                int   oi = __shfl_xor(besti[t][v], mask, 32);
                bool take = (od < bestd[t][v]) || (od == bestd[t][v] && oi < besti[t][v]);
                bestd[t][v] = take ? od : bestd[t][v];
                besti[t][v] = take ? oi : besti[t][v];
            }

    // ---- epilogue: writer lanes emit z_q (== z_q_st forward value), index, loss partial ----
    float lsum = 0.0f;
#pragma unroll
    for (int t = 0; t < RT_PER_WAVE; ++t) {
#pragma unroll
        for (int v = 0; v < 8; ++v) {
            if (m16 == v) {                      // lanes v and 16+v handle rows v and v+8
                int row = v + 8 * (lane >> 4);
                int r   = wave_row_base + t * 16 + row;
                int i   = besti[t][v];
                float4 e = ((const float4*)cb)[i];
                int bb = r >> 14;
                int mm = r & (CH_STRIDE - 1);
                float*       op = out + bb * B_STRIDE + mm;
                const float* zp = z   + bb * B_STRIDE + mm;
                float ec[4] = {e.x, e.y, e.z, e.w};
#pragma unroll
                for (int c = 0; c < 4; ++c) {
                    op[c * CH_STRIDE] = ec[c];
                    float dv = ec[c] - zp[c * CH_STRIDE];
                    lsum += dv * dv;
                }
                out[OUT_IDX_OFF + r] = (float)i;
            }
        }
    }

    // deterministic block-level loss reduction
    lds_red[tid] = lsum;
    __syncthreads();
    for (int s = 128; s > 0; s >>= 1) {
        if (tid < s) lds_red[tid] += lds_red[tid + s];
        __syncthreads();
    }
    if (tid == 0) partial[blockIdx.x] = lds_red[0];
}

__global__ __launch_bounds__(256) void vq_loss(const float* __restrict__ partial,
                                               float* __restrict__ out)
{
    __shared__ float red[256];
    int tid = threadIdx.x;
    red[tid] = partial[tid];   // exactly 256 block partials
    __syncthreads();
    for (int s = 128; s > 0; s >>= 1) {
        if (tid < s) red[tid] += red[tid + s];
        __syncthreads();
    }
    // codebook_loss + 0.25*commitment: both equal mean((z_q - z)^2) numerically
    if (tid == 0) out[OUT_LOSS_OFF] = 1.25f * red[0] / (float)(N_ROWS * CDIM);
}

extern "C" void kernel_launch(void* const* d_in, const int* in_sizes, int n_in,
                              void* d_out, int out_size, void* d_ws, size_t ws_size,
                              hipStream_t stream) {
    const float* z  = (const float*)d_in[0];   // (4,4,8,32,64) f32
    const float* cb = (const float*)d_in[1];   // (8192,4) f32
    float* out      = (float*)d_out;
    float* partial  = (float*)d_ws;            // 256 floats of scratch

    vq_main<<<N_ROWS / ROWS_PER_WG, 256, 0, stream>>>(z, cb, out, partial);
    vq_loss<<<1, 256, 0, stream>>>(partial, out);
}